// quant_discretizer_55413668053129
// MI455X (gfx1250) — compile-verified
//
#include <hip/hip_runtime.h>
#include <cstdint>
#include <cstddef>

// Problem constants (from reference with x=(32,16384,16), coords=(32,2,16384))
#define B_    32      // batch
#define N_    16384   // points per batch
#define NF_   16      // features
#define NQ_   64      // n_q (quantile buckets), also NB (level-1 buckets)
#define M1_   256     // N_/NQ_ elements per level-1 bucket
#define M2_   4       // M1_/NQ_ elements per 2-D cell
#define P_    66      // padded grid 64+2
#define HI_   132     // P_*f, f=2
#define CH_   64      // m2*nf channels
#define NOUT_ 64      // output grid

// GEMM padding: M 66->80 (5 tiles), N 132->144 (9 tiles), K 66->68 (17 steps)
#define MP_   80
#define NP_   144
#define KP_   68

typedef float v2f __attribute__((ext_vector_type(2)));
typedef float v8f __attribute__((ext_vector_type(8)));

// ---------------------------------------------------------------------------
// K0: precompute bicubic tap tables (idx,w per output col) and the padded,
//     K-contiguous weight matrix WwPad[n][k] (144 x 68, zero-padded):
//     WwPad[n][k] = bicubic weight of input row k for output col n.
//     Single block.
// ---------------------------------------------------------------------------
__global__ void qd_k0_tables(float* __restrict__ WwPad, int* __restrict__ cubIdx,
                             float* __restrict__ cubW) {
  int t = threadIdx.x;
  for (int i = t; i < NP_ * KP_; i += blockDim.x) WwPad[i] = 0.f;
  __syncthreads();
  if (t < HI_) {
    const float A = -0.75f;
    float src = (float)t * ((float)(P_ - 1) / (float)(HI_ - 1));
    float i0f = floorf(src);
    int i0 = (int)i0f;
    float x = src - i0f;
    auto c1f = [&](float v) { return ((A + 2.f) * v - (A + 3.f)) * v * v + 1.f; };
    auto c2f = [&](float v) { return ((A * v - 5.f * A) * v + 8.f * A) * v - 4.f * A; };
    float w[4] = { c2f(x + 1.f), c1f(x), c1f(1.f - x), c2f(2.f - x) };
    for (int j = 0; j < 4; j++) {
      int ii = i0 - 1 + j;
      ii = ii < 0 ? 0 : (ii > P_ - 1 ? P_ - 1 : ii);
      cubIdx[t * 4 + j] = ii;
      cubW[t * 4 + j] = w[j];
    }
    for (int j = 0; j < 4; j++)  // thread t owns row t of WwPad: race-free
      WwPad[t * KP_ + cubIdx[t * 4 + j]] += cubW[t * 4 + j];
  }
}

// ---------------------------------------------------------------------------
// K1: level-1 bucketize. One block per batch. Bitonic-sort 16384 values in
//     LDS -> 63 quantile thresholds -> stable counting sort by bucket
//     (== jnp.argsort(buckets, stable)). Dynamic LDS (~82 KB of 320 KB WGP).
// ---------------------------------------------------------------------------
__global__ void qd_k1_bucket1(const float* __restrict__ coords,
                              int* __restrict__ indices1) {
  extern __shared__ unsigned char smem[];
  float* s     = (float*)smem;          // N_ floats (sort buffer)
  float* thr   = s + N_;                // 64
  int*   hist  = (int*)(thr + 64);      // 64 chunks x 64 buckets
  int*   bstart= hist + 64 * 64;        // 64
  int*   bkt   = bstart + 64;           // 256
  int b = blockIdx.x, tid = threadIdx.x;
  const float* c0 = coords + (size_t)b * 2 * N_;

  for (int i = tid; i < N_; i += 256) s[i] = c0[i];
  __syncthreads();
  for (unsigned k = 2; k <= (unsigned)N_; k <<= 1) {
    for (unsigned j = k >> 1; j > 0; j >>= 1) {
      for (unsigned i = tid; i < (unsigned)N_; i += 256) {
        unsigned ixj = i ^ j;
        if (ixj > i) {
          bool up = ((i & k) == 0);
          float a = s[i], bb = s[ixj];
          if ((a > bb) == up) { s[i] = bb; s[ixj] = a; }
        }
      }
      __syncthreads();
    }
  }
  if (tid < NQ_ - 1) {  // jnp.quantile, method=linear: pos = q*(n-1)
    float pos = (float)(tid + 1) * (float)(N_ - 1) / (float)NQ_;
    int i0 = (int)pos;
    float fr = pos - (float)i0;
    thr[tid] = s[i0] + (s[i0 + 1] - s[i0]) * fr;
  }
  __syncthreads();
  for (int i = tid; i < 64 * 64; i += 256) hist[i] = 0;
  __syncthreads();
  for (int i = tid; i < N_; i += 256) {
    float v = c0[i];
    int bu = 0;
    for (int q = 0; q < NQ_ - 1; q++) bu += (thr[q] <= v) ? 1 : 0;
    atomicAdd(&hist[((i >> 8) << 6) + bu], 1);
  }
  __syncthreads();
  if (tid < 64) {  // per-bucket exclusive prefix over chunks (in place)
    int run = 0;
    for (int ch = 0; ch < 64; ch++) {
      int idx = (ch << 6) + tid;
      int tmp = hist[idx];
      hist[idx] = run;
      run += tmp;
    }
    bstart[tid] = run;  // total count per bucket
  }
  __syncthreads();
  if (tid == 0) {  // exclusive prefix over buckets
    int ex = 0;
    for (int q = 0; q < 64; q++) { int tmp = bstart[q]; bstart[q] = ex; ex += tmp; }
  }
  __syncthreads();
  for (int base = 0; base < N_; base += 256) {
    int i = base + tid;
    float v = c0[i];
    int bu = 0;
    for (int q = 0; q < NQ_ - 1; q++) bu += (thr[q] <= v) ? 1 : 0;
    bkt[tid] = bu;
    __syncthreads();
    int rank = 0;  // stable intra-chunk rank
    for (int j = 0; j < tid; j++) rank += (bkt[j] == bu) ? 1 : 0;
    int pos = bstart[bu] + hist[((base >> 8) << 6) + bu] + rank;
    indices1[(size_t)b * N_ + pos] = i;
    __syncthreads();
  }
}

// ---------------------------------------------------------------------------
// K2: level-2 bucketize. One block per (batch, level-1 bucket) row of 256.
// ---------------------------------------------------------------------------
__global__ void qd_k2_bucket2(const float* __restrict__ coords,
                              const int* __restrict__ indices1,
                              int* __restrict__ indicesTot) {
  __shared__ float vals[256], srt[256], thr[64];
  __shared__ int hist[64], bkt[256];
  int blk = blockIdx.x;
  int b = blk >> 6, r = blk & 63, t = threadIdx.x;
  const float* c1 = coords + (size_t)b * 2 * N_ + N_;
  int gi = indices1[(size_t)b * N_ + r * 256 + t];
  float v = c1[gi];
  vals[t] = v;
  srt[t] = v;
  __syncthreads();
  for (unsigned k = 2; k <= 256; k <<= 1) {
    for (unsigned j = k >> 1; j > 0; j >>= 1) {
      unsigned i = t, ixj = i ^ j;
      if (ixj > i) {
        bool up = ((i & k) == 0);
        float a = srt[i], bb = srt[ixj];
        if ((a > bb) == up) { srt[i] = bb; srt[ixj] = a; }
      }
      __syncthreads();
    }
  }
  if (t < 63) {
    float pos = (float)(t + 1) * 255.f / 64.f;
    int i0 = (int)pos;
    float fr = pos - (float)i0;
    thr[t] = srt[i0] + (srt[i0 + 1] - srt[i0]) * fr;
  }
  if (t < 64) hist[t] = 0;
  __syncthreads();
  int bu = 0;
  for (int q = 0; q < 63; q++) bu += (thr[q] <= v) ? 1 : 0;
  bkt[t] = bu;
  atomicAdd(&hist[bu], 1);
  __syncthreads();
  if (t == 0) {
    int ex = 0;
    for (int q = 0; q < 64; q++) { int tmp = hist[q]; hist[q] = ex; ex += tmp; }
  }
  __syncthreads();
  int rank = 0;
  for (int j = 0; j < t; j++) rank += (bkt[j] == bu) ? 1 : 0;
  int pos2 = hist[bu] + rank;
  indicesTot[(size_t)blk * 256 + pos2] = gi;
}

// ---------------------------------------------------------------------------
// K3: cm = per-cell 2nd-smallest of the 4 coord values (each dim).
// ---------------------------------------------------------------------------
__global__ void qd_k3_cm(const float* __restrict__ coords,
                         const int* __restrict__ indicesTot,
                         float* __restrict__ cm) {
  int idx = blockIdx.x * blockDim.x + threadIdx.x;
  if (idx >= B_ * 64 * 64) return;
  int q = idx & 63, r = (idx >> 6) & 63, b = idx >> 12;
  const int* it = indicesTot + ((size_t)(b * 64 + r)) * 256 + q * 4;
  int i0 = it[0], i1 = it[1], i2 = it[2], i3 = it[3];
  const float* cb = coords + (size_t)b * 2 * N_;
#pragma unroll
  for (int d = 0; d < 2; d++) {
    const float* cd = cb + d * N_;
    float a0 = cd[i0], a1 = cd[i1], a2 = cd[i2], a3 = cd[i3], tt;
    if (a0 > a1) { tt = a0; a0 = a1; a1 = tt; }
    if (a2 > a3) { tt = a2; a2 = a3; a3 = tt; }
    if (a0 > a2) { tt = a0; a0 = a2; a2 = tt; }
    if (a1 > a3) { tt = a1; a1 = a3; a3 = tt; }
    if (a1 > a2) { tt = a1; a1 = a2; a2 = tt; }
    cm[(size_t)b * 2 * 4096 + (size_t)d * 4096 + r * 64 + q] = a1;
  }
}

// ---------------------------------------------------------------------------
// K4: scale_coords2 -> padded (b, 2, 66, 66). One block per batch.
// ---------------------------------------------------------------------------
__global__ void qd_k4_scale(const float* __restrict__ cm,
                            float* __restrict__ scaled) {
  __shared__ float mn0[64], mx0[64], mn1[64], mx1[64];
  int b = blockIdx.x, t = threadIdx.x;
  const float* c0 = cm + (size_t)b * 2 * 4096;
  const float* c1 = c0 + 4096;
  if (t < 64) {  // column stats of c0 (min/max over axis=1)
    float mn = 3.4e38f, mx = -3.4e38f;
    for (int i = 0; i < 64; i++) { float v = c0[i * 64 + t]; mn = fminf(mn, v); mx = fmaxf(mx, v); }
    mn0[t] = mn; mx0[t] = mx;
  } else if (t < 128) {  // row stats of c1 (min/max over axis=-1)
    int i = t - 64;
    float mn = 3.4e38f, mx = -3.4e38f;
    for (int j = 0; j < 64; j++) { float v = c1[i * 64 + j]; mn = fminf(mn, v); mx = fmaxf(mx, v); }
    mn1[i] = mn; mx1[i] = mx;
  }
  __syncthreads();
  float* s0 = scaled + (size_t)b * 2 * P_ * P_;
  float* s1 = s0 + P_ * P_;
  for (int cell = t; cell < P_ * P_; cell += blockDim.x) {
    int I = cell / P_, J = cell % P_;
    int jc = J - 1; jc = jc < 0 ? 0 : (jc > 63 ? 63 : jc);
    int ic = I - 1; ic = ic < 0 ? 0 : (ic > 63 ? 63 : ic);
    float v0;
    if (I == 0)            v0 = mn0[jc] - 0.5f;
    else if (I == P_ - 1)  v0 = mx0[jc] + 0.5f;
    else                   v0 = c0[(I - 1) * 64 + jc];
    s0[cell] = v0;
    float v1;
    if (J == 0)            v1 = mn1[ic] - 0.5f;
    else if (J == P_ - 1)  v1 = mx1[ic] + 0.5f;
    else                   v1 = c1[ic * 64 + (J - 1)];
    s1[cell] = v1;
  }
}

// ---------------------------------------------------------------------------
// K5: gather + edge-pad data -> dataPad (b, 66, 66, 64ch), channel-fastest.
// ---------------------------------------------------------------------------
__global__ void qd_k5_datapad(const float* __restrict__ x,
                              const int* __restrict__ indicesTot,
                              float* __restrict__ dataPad) {
  long idx = (long)blockIdx.x * blockDim.x + threadIdx.x;
  const long total = (long)B_ * P_ * P_ * CH_;
  if (idx >= total) return;
  int c = (int)(idx & 63);
  long rem = idx >> 6;
  int J = (int)(rem % P_); rem /= P_;
  int I = (int)(rem % P_);
  int b = (int)(rem / P_);
  int i = I - 1; i = i < 0 ? 0 : (i > 63 ? 63 : i);
  int j = J - 1; j = j < 0 ? 0 : (j > 63 ? 63 : j);
  int m = c >> 4, f = c & 15;
  int gi = indicesTot[((size_t)(b * 64 + i)) * 256 + j * 4 + m];
  dataPad[idx] = x[((size_t)b * N_ + gi) * NF_ + f];
}

// ---------------------------------------------------------------------------
// K6: T2[b,c] (80x144, padded) = X[b,c] (66x66) * Ww^T via f32 WMMA 16x16x4.
//     One block per (b,c). Both operands staged zero-padded in LDS with
//     K-contiguous layout so every fragment is a single unconditional
//     ds_load_b64; D tiles are stored unconditionally to the padded T2
//     (no bounds checks anywhere in the WMMA path -> no EXEC juggling).
//     8 waves cover the 5x9 output tiles; EXEC is all-ones inside each wave.
// ---------------------------------------------------------------------------
__global__ void qd_k6_gemm(const float* __restrict__ dataPad,
                           const float* __restrict__ WwPad,
                           float* __restrict__ T2) {
  __shared__ float Xs[MP_ * KP_];   // 80 x 68, zero-padded     (21.8 KB)
  __shared__ float Wws[NP_ * KP_];  // 144 x 68, K-contiguous   (39.2 KB)
  int blk = blockIdx.x;
  int b = blk >> 6, c = blk & 63;
  int tid = threadIdx.x;
  for (int i = tid; i < MP_ * KP_; i += 256) Xs[i] = 0.f;
  for (int i = tid; i < NP_ * KP_; i += 256) Wws[i] = WwPad[i];
  __syncthreads();
  const float* dp = dataPad + ((size_t)b * P_ * P_) * CH_ + c;
  for (int e = tid; e < P_ * P_; e += 256) {
    int i = e / P_, j = e % P_;
    Xs[i * KP_ + j] = dp[(size_t)e * CH_];
  }
  __syncthreads();
  float* dst = T2 + (size_t)blk * MP_ * NP_;
  __builtin_prefetch(dst, 1, 1);  // global_prefetch_b8
#if defined(__gfx1250__) && __has_builtin(__builtin_amdgcn_wmma_f32_16x16x4_f32)
  int wv = tid >> 5, lane = tid & 31, half = lane >> 4, l16 = lane & 15;
  for (int t = wv; t < 5 * 9; t += 8) {  // wave-granular tile distribution
    int mt = t / 9, nt = t % 9;
    v8f acc = {0.f, 0.f, 0.f, 0.f, 0.f, 0.f, 0.f, 0.f};
    int m = mt * 16 + l16;  // A lane row (ISA 7.12.2: 32-bit A 16x4)
    int n = nt * 16 + l16;  // B/D lane col
    const float* arow = &Xs[m * KP_];
    const float* brow = &Wws[n * KP_];
    for (int k0 = 0; k0 < KP_; k0 += 4) {
      int kk = k0 + 2 * half;  // K = vgpr + 2*(lane>=16); 8B-aligned (kk even)
      v2f a  = *(const v2f*)(arow + kk);  // one ds_load_b64
      v2f bb = *(const v2f*)(brow + kk);  // one ds_load_b64
      acc = __builtin_amdgcn_wmma_f32_16x16x4_f32(false, a, false, bb,
                                                  (short)0, acc, false, false);
    }
#pragma unroll
    for (int r = 0; r < 8; r++) {  // C/D: VGPR r -> M = r + 8*(lane>=16)
      int row = mt * 16 + r + half * 8;
      dst[row * NP_ + n] = acc[r];  // unconditional: T2 is tile-padded
    }
  }
#else
  // Scalar fallback (host pass / builtin unavailable)
  for (int e = tid; e < MP_ * NP_; e += 256) {
    int row = e / NP_, col = e % NP_;
    float sum = 0.f;
    for (int k = 0; k < KP_; k++) sum += Xs[row * KP_ + k] * Wws[col * KP_ + k];
    dst[e] = sum;
  }
#endif
}

// ---------------------------------------------------------------------------
// K7: bilinear 66 -> 132 of both coord planes.
// ---------------------------------------------------------------------------
__global__ void qd_k7_bilinear(const float* __restrict__ scaled,
                               float* __restrict__ ci) {
  long idx = (long)blockIdx.x * blockDim.x + threadIdx.x;
  const long total = (long)B_ * 2 * HI_ * HI_;
  if (idx >= total) return;
  int w = (int)(idx % HI_);
  long rem = idx / HI_;
  int h = (int)(rem % HI_); rem /= HI_;
  int d = (int)(rem % 2);
  int b = (int)(rem / 2);
  const float sc = (float)(P_ - 1) / (float)(HI_ - 1);
  float sh = h * sc, sw = w * sc;
  int h0 = (int)floorf(sh); h0 = h0 < 0 ? 0 : (h0 > P_ - 1 ? P_ - 1 : h0);
  int w0 = (int)floorf(sw); w0 = w0 < 0 ? 0 : (w0 > P_ - 1 ? P_ - 1 : w0);
  int h1 = h0 + 1 > P_ - 1 ? P_ - 1 : h0 + 1;
  int w1 = w0 + 1 > P_ - 1 ? P_ - 1 : w0 + 1;
  float th = sh - (float)h0, tw = sw - (float)w0;
  const float* s = scaled + ((size_t)b * 2 + d) * P_ * P_;
  float v00 = s[h0 * P_ + w0], v01 = s[h0 * P_ + w1];
  float v10 = s[h1 * P_ + w0], v11 = s[h1 * P_ + w1];
  ci[idx] = (1.f - th) * ((1.f - tw) * v00 + tw * v01) +
            th * ((1.f - tw) * v10 + tw * v11);
}

// ---------------------------------------------------------------------------
// K8: argmin over H of |ci0[h,w]-grid[g]| (first min) -> h*; also pick ci1
//     at h* (coords_inter0 channel 1).
// ---------------------------------------------------------------------------
__global__ void qd_k8_argmin_h(const float* __restrict__ ci,
                               int* __restrict__ indexC,
                               float* __restrict__ ci1sel) {
  int idx = blockIdx.x * blockDim.x + threadIdx.x;
  if (idx >= B_ * NOUT_ * HI_) return;
  int w = idx % HI_;
  int rem = idx / HI_;
  int g = rem % NOUT_;
  int b = rem / NOUT_;
  const float* c0p = ci + (size_t)b * 2 * HI_ * HI_;
  float gv = (float)g * (1.f / 63.f);
  float bd = 3.4e38f;
  int bh = 0;
  for (int h = 0; h < HI_; h++) {
    float dd = fabsf(c0p[h * HI_ + w] - gv);
    if (dd < bd) { bd = dd; bh = h; }  // strict < => first occurrence
  }
  indexC[idx] = bh;
  ci1sel[idx] = c0p[HI_ * HI_ + bh * HI_ + w];
}

// ---------------------------------------------------------------------------
// K9: argmin over W of |ci1sel[g,w]-grid[g2]| -> w*.
// ---------------------------------------------------------------------------
__global__ void qd_k9_argmin_w(const float* __restrict__ ci1sel,
                               int* __restrict__ ind2) {
  int idx = blockIdx.x * blockDim.x + threadIdx.x;
  if (idx >= B_ * NOUT_ * NOUT_) return;
  int g2 = idx % NOUT_;
  int rem = idx / NOUT_;
  int g = rem % NOUT_;
  int b = rem / NOUT_;
  const float* row = ci1sel + ((size_t)(b * NOUT_ + g)) * HI_;
  float gv = (float)g2 * (1.f / 63.f);
  float bd = 3.4e38f;
  int bw = 0;
  for (int w = 0; w < HI_; w++) {
    float dd = fabsf(row[w] - gv);
    if (dd < bd) { bd = dd; bw = w; }
  }
  ind2[idx] = bw;
}

// ---------------------------------------------------------------------------
// K10: out[b,c,g2,g] = sum_j cubW[h*,j] * T2[b,c, cubIdx[h*,j], w*]
//      (fused row-direction bicubic at the selected (h*, w*)).
//      T2 rows have padded stride NP_ (144).
// ---------------------------------------------------------------------------
__global__ void qd_k10_out(const float* __restrict__ T2,
                           const int* __restrict__ cubIdx,
                           const float* __restrict__ cubW,
                           const int* __restrict__ indexC,
                           const int* __restrict__ ind2,
                           float* __restrict__ out) {
  long idx = (long)blockIdx.x * blockDim.x + threadIdx.x;
  const long total = (long)B_ * CH_ * NOUT_ * NOUT_;
  if (idx >= total) return;
  int g = (int)(idx % NOUT_);
  long rem = idx / NOUT_;
  int g2 = (int)(rem % NOUT_); rem /= NOUT_;
  int c = (int)(rem % CH_);
  int b = (int)(rem / CH_);
  int w = ind2[((size_t)(b * NOUT_ + g)) * NOUT_ + g2];
  int h = indexC[((size_t)(b * NOUT_ + g)) * HI_ + w];
  const float* t2 = T2 + ((size_t)(b * CH_ + c)) * MP_ * NP_;
  float sum = 0.f;
#pragma unroll
  for (int j = 0; j < 4; j++)
    sum += cubW[h * 4 + j] * t2[cubIdx[h * 4 + j] * NP_ + w];
  out[idx] = sum;
}

// ---------------------------------------------------------------------------
extern "C" void kernel_launch(void* const* d_in, const int* in_sizes, int n_in,
                              void* d_out, int out_size, void* d_ws, size_t ws_size,
                              hipStream_t stream) {
  (void)in_sizes; (void)n_in; (void)out_size; (void)ws_size;
  const float* x      = (const float*)d_in[0];  // (32,16384,16) f32
  const float* coords = (const float*)d_in[1];  // (32,2,16384) f32
  float* out = (float*)d_out;                   // (32,4,16,64,64) f32

  unsigned char* ws = (unsigned char*)d_ws;
  size_t off = 0;
  auto carve = [&](size_t bytes) -> void* {
    void* p = ws + off;
    off = (off + bytes + 255) & ~(size_t)255;
    return p;
  };
  int*   indices1   = (int*)  carve((size_t)B_ * N_ * 4);             // 2 MB
  int*   indicesTot = (int*)  carve((size_t)B_ * 64 * 256 * 4);       // 2 MB
  float* cm         = (float*)carve((size_t)B_ * 2 * 4096 * 4);       // 1 MB
  float* scaled     = (float*)carve((size_t)B_ * 2 * P_ * P_ * 4);
  float* dataPad    = (float*)carve((size_t)B_ * P_ * P_ * CH_ * 4);  // 35.7 MB
  float* T2         = (float*)carve((size_t)B_ * CH_ * MP_ * NP_ * 4);// 94.4 MB (tile-padded)
  float* WwPad      = (float*)carve((size_t)NP_ * KP_ * 4);           // padded Ww
  int*   cubIdx     = (int*)  carve((size_t)HI_ * 4 * 4);
  float* cubW       = (float*)carve((size_t)HI_ * 4 * 4);
  float* ci         = (float*)carve((size_t)B_ * 2 * HI_ * HI_ * 4);  // 4.5 MB
  int*   indexC     = (int*)  carve((size_t)B_ * NOUT_ * HI_ * 4);
  float* ci1sel     = (float*)carve((size_t)B_ * NOUT_ * HI_ * 4);
  int*   ind2       = (int*)  carve((size_t)B_ * NOUT_ * NOUT_ * 4);

  const size_t k1Smem = (size_t)(N_ + 64) * 4 + (size_t)(64 * 64 + 64 + 256) * 4;

  qd_k0_tables<<<1, 256, 0, stream>>>(WwPad, cubIdx, cubW);
  qd_k1_bucket1<<<B_, 256, k1Smem, stream>>>(coords, indices1);
  qd_k2_bucket2<<<B_ * 64, 256, 0, stream>>>(coords, indices1, indicesTot);
  qd_k3_cm<<<(B_ * 4096 + 255) / 256, 256, 0, stream>>>(coords, indicesTot, cm);
  qd_k4_scale<<<B_, 256, 0, stream>>>(cm, scaled);
  {
    long total = (long)B_ * P_ * P_ * CH_;
    qd_k5_datapad<<<(int)((total + 255) / 256), 256, 0, stream>>>(x, indicesTot, dataPad);
  }
  qd_k6_gemm<<<B_ * CH_, 256, 0, stream>>>(dataPad, WwPad, T2);
  {
    long total = (long)B_ * 2 * HI_ * HI_;
    qd_k7_bilinear<<<(int)((total + 255) / 256), 256, 0, stream>>>(scaled, ci);
  }
  qd_k8_argmin_h<<<(B_ * NOUT_ * HI_ + 255) / 256, 256, 0, stream>>>(ci, indexC, ci1sel);
  qd_k9_argmin_w<<<(B_ * NOUT_ * NOUT_ + 255) / 256, 256, 0, stream>>>(ci1sel, ind2);
  {
    long total = (long)B_ * CH_ * NOUT_ * NOUT_;
    qd_k10_out<<<(int)((total + 255) / 256), 256, 0, stream>>>(T2, cubIdx, cubW, indexC, ind2, out);
  }
}